// VectorQuantizerEMA_76811195122172
// MI455X (gfx1250) — compile-verified
//
#include <hip/hip_runtime.h>
#include <math.h>
#include <stdint.h>

typedef __attribute__((ext_vector_type(2))) float f32x2;
typedef __attribute__((ext_vector_type(8))) float f32x8;

#define NROWS  32768      // 32*1024
#define DDIM   256
#define KCODES 1024
#define NCHUNK 64         // KCODES / 16
#define KSTEPS 64         // DDIM / 4
#define COMMITF 0.25f
#define DECAYF  0.99f
#define EPSF    1e-5f

// ---------------------------------------------------------------- init
__global__ void zero_ws_kernel(float* __restrict__ ws, int n) {
  int i = blockIdx.x * blockDim.x + threadIdx.x;
  if (i < n) ws[i] = 0.0f;
}

// e2[k] = ||codebook[k]||^2 ; one block (256 threads) per code row
__global__ void code_norm_kernel(const float* __restrict__ cb, float* __restrict__ e2) {
  const int k = blockIdx.x;
  float v = cb[(size_t)k * DDIM + threadIdx.x];
  v *= v;
  #pragma unroll
  for (int s = 1; s < 32; s <<= 1) v += __shfl_xor(v, s, 32);
  __shared__ float part[8];
  if ((threadIdx.x & 31) == 0) part[threadIdx.x >> 5] = v;
  __syncthreads();
  if (threadIdx.x == 0) {
    float s = 0.f;
    #pragma unroll
    for (int i = 0; i < 8; i++) s += part[i];
    e2[k] = s;
  }
}

// ---------------------------------------------------------------- phase 1
// Async (memory -> LDS) staging of a 16-code chunk into the swizzled WMMA
// fragment layout: chunk kc occupies 64 words, lane l owns words [2l,2l+1].
// Each float4 of a codebook row splits into two contiguous 8B pieces, which
// map onto two GLOBAL_LOAD_ASYNC_TO_LDS_B64 ops (no VGPR round trip,
// tracked by ASYNCcnt). 16 async ops per wave per chunk.
__device__ __forceinline__ void stage_chunk_async(const float* __restrict__ et,
                                                  float* __restrict__ buf,
                                                  int tid) {
  #pragma unroll
  for (int i = 0; i < 8; i++) {
    int flat = i * 128 + tid;          // 0..1023 float4s in the 16x256 chunk
    int m  = flat >> 6;                // code row in chunk
    int k4 = flat & 63;                // k-chunk index
    const float* g0 = et + m * DDIM + k4 * 4;
    uint32_t l0 = (uint32_t)(uintptr_t)(buf + k4 * 64 + 2 * m);       // k%4=0,1
    uint32_t l1 = (uint32_t)(uintptr_t)(buf + k4 * 64 + 32 + 2 * m);  // k%4=2,3
    asm volatile("global_load_async_to_lds_b64 %0, %1, off"
                 :: "v"(l0), "v"((uint64_t)(uintptr_t)g0) : "memory");
    asm volatile("global_load_async_to_lds_b64 %0, %1, off"
                 :: "v"(l1), "v"((uint64_t)(uintptr_t)(g0 + 2)) : "memory");
  }
}

// 4 waves/block; each wave computes argmin over all 1024 codes for 16 rows
// using v_wmma_f32_16x16x4_f32 with 4 independent accumulators (breaks the
// per-WMMA RAW chain) and double-buffered async codebook staging.
__global__ void __launch_bounds__(128)
argmin_wmma_kernel(const float* __restrict__ z, const float* __restrict__ cb,
                   const float* __restrict__ e2, float* __restrict__ counts,
                   int* __restrict__ idxi, float* __restrict__ o_idx,
                   float* __restrict__ lacc) {
  __shared__ float zbuf[4 * 4096];   // 4 waves x (16 rows x 256 k), frag order
  __shared__ float ebuf[2 * 4096];   // double-buffered 16-code chunk
  __shared__ float s_loss;

  const int tid  = threadIdx.x;
  const int lane = tid & 31;
  const int wid  = tid >> 5;
  const int rbase = blockIdx.x * 64 + wid * 16;

  if (tid == 0) s_loss = 0.0f;

  // kick off async fill of chunk 0 so it overlaps the z staging below
  stage_chunk_async(cb, ebuf, tid);

  // stage this wave's 16x256 z tile into swizzled LDS fragments
  float* zb = zbuf + wid * 4096;
  {
    const float* zt = z + (size_t)rbase * DDIM;
    #pragma unroll
    for (int i = 0; i < 32; i++) {
      int flat = i * 32 + lane;         // 0..1023 float4s
      int m  = flat >> 6;               // row in tile
      int k4 = flat & 63;               // k-chunk index
      const float4 g = *(const float4*)(zt + m * DDIM + k4 * 4);
      f32x2 lo; lo[0] = g.x; lo[1] = g.y;
      f32x2 hi; hi[0] = g.z; hi[1] = g.w;
      *(f32x2*)(zb + k4 * 64 + 2 * m)      = lo;   // k%4 = 0,1 -> lanes 0..15
      *(f32x2*)(zb + k4 * 64 + 32 + 2 * m) = hi;   // k%4 = 2,3 -> lanes 16..31
    }
  }

  float best[8]; int bidx[8];
  #pragma unroll
  for (int r = 0; r < 8; r++) { best[r] = 3.4e38f; bidx[r] = 0; }

  for (int c = 0; c < NCHUNK; c++) {
    float* cur = ebuf + (c & 1) * 4096;
    float* nxt = ebuf + ((c + 1) & 1) * 4096;

    if (c < NCHUNK - 1) {
      // issue next chunk, then drain the current chunk's 16 in-order asyncs
      stage_chunk_async(cb + (size_t)(c + 1) * 16 * DDIM, nxt, tid);
      asm volatile("s_wait_asynccnt 16" ::: "memory");
    } else {
      asm volatile("s_wait_asynccnt 0" ::: "memory");
    }
    __syncthreads();                   // chunk c visible to all waves

    f32x8 acc[4] = {{}, {}, {}, {}};
    #pragma unroll
    for (int kc = 0; kc < KSTEPS; kc += 4) {
      #pragma unroll
      for (int u = 0; u < 4; u++) {
        f32x2 a = *(const f32x2*)(zb  + (kc + u) * 64 + 2 * lane);
        f32x2 b = *(const f32x2*)(cur + (kc + u) * 64 + 2 * lane);
        acc[u] = __builtin_amdgcn_wmma_f32_16x16x4_f32(false, a, false, b,
                                                       (short)0, acc[u],
                                                       false, false);
      }
    }

    const int   ci  = c * 16 + (lane & 15);
    const float e2v = e2[ci];
    #pragma unroll
    for (int r = 0; r < 8; r++) {
      float dot = acc[0][r] + acc[1][r] + acc[2][r] + acc[3][r];
      float sc  = e2v - 2.0f * dot;    // dist - ||z||^2 (row-constant dropped)
      if (sc < best[r]) { best[r] = sc; bidx[r] = ci; }
    }
    __syncthreads();                   // all readers done before nxt overwrite
  }

  // reduce over the 16 lanes holding the code classes of each row
  #pragma unroll
  for (int r = 0; r < 8; r++) {
    float b = best[r]; int ix = bidx[r];
    #pragma unroll
    for (int s = 1; s < 16; s <<= 1) {
      float ob = __shfl_xor(b, s, 32);
      int   oi = __shfl_xor(ix, s, 32);
      if (ob < b || (ob == b && oi < ix)) { b = ob; ix = oi; }
    }
    if ((lane & 15) == 0) {
      int row = rbase + r + 8 * (lane >> 4);
      idxi[row]  = ix;
      o_idx[row] = (float)ix;
      atomicAdd(&counts[ix], 1.0f);
      atomicAdd(&s_loss, b);           // sum of (dist - ||z||^2)
    }
  }
  __syncthreads();
  if (tid == 0) atomicAdd(lacc, s_loss);
}

// ---------------------------------------------------------------- phase 2
// z_q = codebook[idx] (STE forward value), dw scatter, sum z^2 into loss acc
__global__ void gather_scatter_kernel(const float* __restrict__ z,
                                      const float* __restrict__ cb,
                                      const int* __restrict__ idxi,
                                      float* __restrict__ zq,
                                      float* __restrict__ dw,
                                      float* __restrict__ lacc) {
  const int row = blockIdx.x;
  const int j   = threadIdx.x;
  const int c   = idxi[row];
  const float zv = z[(size_t)row * DDIM + j];
  zq[(size_t)row * DDIM + j] = cb[(size_t)c * DDIM + j];
  atomicAdd(&dw[(size_t)c * DDIM + j], zv);

  float v = zv * zv;
  #pragma unroll
  for (int s = 1; s < 32; s <<= 1) v += __shfl_xor(v, s, 32);
  __shared__ float part[8];
  if ((j & 31) == 0) part[j >> 5] = v;
  __syncthreads();
  if (j == 0) {
    float s = 0.f;
    #pragma unroll
    for (int i = 0; i < 8; i++) s += part[i];
    atomicAdd(lacc, s);
  }
}

// ---------------------------------------------------------------- phase 3
__global__ void finalize_scalars_kernel(const float* __restrict__ counts,
                                        const float* __restrict__ ema_cs,
                                        const float* __restrict__ lacc,
                                        float* __restrict__ o_cs,
                                        float* __restrict__ o_loss,
                                        float* __restrict__ o_perp,
                                        float* __restrict__ nt_ws) {
  const int tid = threadIdx.x;
  float nt = 0.f, pp = 0.f;
  for (int k = tid; k < KCODES; k += 256) {
    float cnt = counts[k];
    float cs  = DECAYF * ema_cs[k] + (1.0f - DECAYF) * cnt;
    o_cs[k] = cs;
    nt += cs;
    float p = cnt * (1.0f / (float)NROWS);
    pp += p * logf(p + 1e-10f);
  }
  #pragma unroll
  for (int s = 1; s < 32; s <<= 1) { nt += __shfl_xor(nt, s, 32); pp += __shfl_xor(pp, s, 32); }
  __shared__ float pn[8], pe[8];
  if ((tid & 31) == 0) { pn[tid >> 5] = nt; pe[tid >> 5] = pp; }
  __syncthreads();
  if (tid == 0) {
    float a = 0.f, b = 0.f;
    #pragma unroll
    for (int i = 0; i < 8; i++) { a += pn[i]; b += pe[i]; }
    nt_ws[0]  = a;
    o_perp[0] = expf(-b);
    o_loss[0] = COMMITF * lacc[0] / (float)((long long)NROWS * DDIM);
  }
}

__global__ void finalize_embed_kernel(const float* __restrict__ ema_embed,
                                      const float* __restrict__ dw,
                                      const float* __restrict__ o_cs,
                                      const float* __restrict__ nt_ws,
                                      float* __restrict__ o_ema,
                                      float* __restrict__ o_cb) {
  const int e = blockIdx.x * 256 + threadIdx.x;
  const int k = e >> 8;                // / DDIM
  float ne = DECAYF * ema_embed[e] + (1.0f - DECAYF) * dw[e];
  o_ema[e] = ne;
  float cs = o_cs[k];
  float nt = nt_ws[0];
  float sm = (cs + EPSF) / (nt + (float)KCODES * EPSF) * nt;
  o_cb[e] = ne / sm;
}

// ---------------------------------------------------------------- launcher
extern "C" void kernel_launch(void* const* d_in, const int* in_sizes, int n_in,
                              void* d_out, int out_size, void* d_ws, size_t ws_size,
                              hipStream_t stream) {
  const float* z   = (const float*)d_in[0];   // [32,1024,256]
  const float* cb  = (const float*)d_in[1];   // [1024,256]
  const float* ecs = (const float*)d_in[2];   // [1024]
  const float* eem = (const float*)d_in[3];   // [1024,256]

  float* out    = (float*)d_out;
  float* o_zq   = out;                        // 8388608
  float* o_loss = out + 8388608;              // 1
  float* o_idx  = out + 8388609;              // 32768 (as float)
  float* o_perp = out + 8421377;              // 1
  float* o_cb   = out + 8421378;              // 262144
  float* o_cs   = out + 8683522;              // 1024
  float* o_ema  = out + 8684546;              // 262144

  float* wsf    = (float*)d_ws;
  float* counts = wsf;                        // 1024
  float* dw     = wsf + 1024;                 // 262144
  float* e2     = wsf + 263168;               // 1024
  float* nt     = wsf + 264192;               // 1
  float* lacc   = wsf + 264193;               // 1
  int*   idxi   = (int*)(wsf + 264194);       // 32768

  const int nzero = 264194;
  zero_ws_kernel<<<(nzero + 255) / 256, 256, 0, stream>>>(wsf, nzero);
  code_norm_kernel<<<KCODES, 256, 0, stream>>>(cb, e2);
  argmin_wmma_kernel<<<NROWS / 64, 128, 0, stream>>>(z, cb, e2, counts, idxi, o_idx, lacc);
  gather_scatter_kernel<<<NROWS, 256, 0, stream>>>(z, cb, idxi, o_zq, dw, lacc);
  finalize_scalars_kernel<<<1, 256, 0, stream>>>(counts, ecs, lacc, o_cs, o_loss, o_perp, nt);
  finalize_embed_kernel<<<(KCODES * DDIM) / 256, 256, 0, stream>>>(eem, dw, o_cs, nt, o_ema, o_cb);
}